// VectorQuantizer_26551487824069
// MI455X (gfx1250) — compile-verified
//
#include <hip/hip_runtime.h>
#include <hip/hip_bf16.h>
#include <math.h>

// ---------------------------------------------------------------------------
// VectorQuantizer forward for MI455X (gfx1250), fp32 WMMA path.
//   latent  : [4,2048,256] f32  -> N_TOK = 8192 rows of D=256
//   codebook: [4096,256]   f32  -> K = 4096 codes
// Outputs (flat f32, in return order):
//   quantized [8192*256], indices [8192], commitment, codebook_loss,
//   perplexity, selected_cosine_sim, avg_euclidean, min_euclidean
// ---------------------------------------------------------------------------

typedef __attribute__((ext_vector_type(2))) float v2f;
typedef __attribute__((ext_vector_type(8))) float v8f;

#define KCODES 4096
#define DDIM   256
#define N_TOK  8192
#define LDSTRIDE 260   // LDS tile stride (260 mod 64 = 4 -> conflict-free)

#define AROWS    32    // latent rows per block in vq_assign
#define ATHREADS 512   // 16 waves
#define AWAVES   16

// ---------------------------------------------------------------------------
__global__ void vq_init(unsigned* __restrict__ counts, double* __restrict__ scal,
                        unsigned* __restrict__ minbits) {
  int t = blockIdx.x * blockDim.x + threadIdx.x;
  if (t < KCODES) counts[t] = 0u;
  if (t == 0) {
    scal[0] = 0.0;  // sum (latent - q)^2
    scal[1] = 0.0;  // sum selected cosine
    scal[2] = 0.0;  // sum off-diagonal pairwise distance
    minbits[0] = 0x7F800000u;  // +inf (uint order == float order for d >= 0)
  }
}

// ---------------------------------------------------------------------------
// Row L2-normalize; one wave per row. Optionally emit raw sum-of-squares.
__global__ __launch_bounds__(256)
void vq_rownorm(const float* __restrict__ in, float* __restrict__ out,
                float* __restrict__ sq, int rows) {
  const int wid  = threadIdx.x >> 5;
  const int lane = threadIdx.x & 31;
  const int row  = blockIdx.x * 8 + wid;
  if (row >= rows) return;
  const float* r = in + (size_t)row * DDIM;
  float vals[8];
  float s = 0.f;
#pragma unroll
  for (int i = 0; i < 8; ++i) { vals[i] = r[lane + 32 * i]; s += vals[i] * vals[i]; }
#pragma unroll
  for (int off = 16; off >= 1; off >>= 1) s += __shfl_xor(s, off, 32);
  if (sq != nullptr && lane == 0) sq[row] = s;
  const float inv = 1.0f / fmaxf(sqrtf(s), 1e-12f);
  float* o = out + (size_t)row * DDIM;
#pragma unroll
  for (int i = 0; i < 8; ++i) o[lane + 32 * i] = vals[i] * inv;
}

// ---------------------------------------------------------------------------
// Main kernel: 32 latent rows per block (16 waves). Each wave sweeps 16-code
// tiles; each B fragment load feeds TWO independent WMMA accumulators
// (rows 0-15 and 16-31), halving codebook traffic per MAC.
__global__ __launch_bounds__(ATHREADS)
void vq_assign(const float* __restrict__ latent, const float* __restrict__ codebook,
               const float* __restrict__ ln, const float* __restrict__ cn,
               unsigned* __restrict__ counts,
               float* __restrict__ out_q, float* __restrict__ out_idx,
               double* __restrict__ scal) {
  __shared__ float lA[AROWS * LDSTRIDE];   // 33,280 B
  __shared__ float s_bv[AWAVES * AROWS];
  __shared__ int   s_bi[AWAVES * AROWS];
  __shared__ int   s_idx[AROWS];
  __shared__ float s_val[AROWS];
  __shared__ float s_red[ATHREADS];

  const int tid  = threadIdx.x;
  const int wid  = tid >> 5;
  const int lane = tid & 31;
  const int half = lane >> 4;   // K-pair select within fragment
  const int l16  = lane & 15;   // M index (A) / N index (B)
  const int r0   = blockIdx.x * AROWS;

  // Stage normalized-latent tile (32 x 256) into LDS.
  for (int i = tid; i < AROWS * DDIM; i += ATHREADS) {
    const int row = i >> 8;
    const int col = i & 255;
    lA[row * LDSTRIDE + col] = ln[(size_t)(r0 + row) * DDIM + col];
  }
  __syncthreads();

  const float* a0 = &lA[l16 * LDSTRIDE + 2 * half];
  const float* a1 = a0 + 16 * LDSTRIDE;

  float bv0[8], bv1[8];
  int   bi0[8], bi1[8];
#pragma unroll
  for (int j = 0; j < 8; ++j) {
    bv0[j] = -1e30f; bi0[j] = 0x7FFFFFFF;
    bv1[j] = -1e30f; bi1[j] = 0x7FFFFFFF;
  }

  for (int kt = wid; kt < KCODES / 16; kt += AWAVES) {
    const int c0 = kt * 16;
    const float* bBase = cn + (size_t)(c0 + l16) * DDIM + 2 * half;
    v8f acc0 = {};
    v8f acc1 = {};
#pragma unroll 8
    for (int kk = 0; kk < 64; ++kk) {
      v2f b  = *(const v2f*)(bBase + 4 * kk);
      v2f x0 = *(const v2f*)(a0 + 4 * kk);
      v2f x1 = *(const v2f*)(a1 + 4 * kk);
      acc0 = __builtin_amdgcn_wmma_f32_16x16x4_f32(false, x0, false, b, (short)0,
                                                   acc0, false, false);
      acc1 = __builtin_amdgcn_wmma_f32_16x16x4_f32(false, x1, false, b, (short)0,
                                                   acc1, false, false);
    }
    const int idx = c0 + l16;
#pragma unroll
    for (int j = 0; j < 8; ++j) {
      const float v0 = acc0[j];
      if (v0 > bv0[j] || (v0 == bv0[j] && idx < bi0[j])) { bv0[j] = v0; bi0[j] = idx; }
      const float v1 = acc1[j];
      if (v1 > bv1[j] || (v1 == bv1[j] && idx < bi1[j])) { bv1[j] = v1; bi1[j] = idx; }
    }
  }

  // Reduce over the 16 lanes of each half (each half owns rows j / j+8).
#pragma unroll
  for (int j = 0; j < 8; ++j) {
    float v = bv0[j]; int idx = bi0[j];
#pragma unroll
    for (int off = 8; off >= 1; off >>= 1) {
      const float ov = __shfl_xor(v, off, 32);
      const int   oi = __shfl_xor(idx, off, 32);
      if (ov > v || (ov == v && oi < idx)) { v = ov; idx = oi; }
    }
    bv0[j] = v; bi0[j] = idx;

    v = bv1[j]; idx = bi1[j];
#pragma unroll
    for (int off = 8; off >= 1; off >>= 1) {
      const float ov = __shfl_xor(v, off, 32);
      const int   oi = __shfl_xor(idx, off, 32);
      if (ov > v || (ov == v && oi < idx)) { v = ov; idx = oi; }
    }
    bv1[j] = v; bi1[j] = idx;
  }
  if (l16 == 0) {
#pragma unroll
    for (int j = 0; j < 8; ++j) {
      const int row0 = j + 8 * half;
      s_bv[wid * AROWS + row0] = bv0[j];
      s_bi[wid * AROWS + row0] = bi0[j];
      const int row1 = 16 + j + 8 * half;
      s_bv[wid * AROWS + row1] = bv1[j];
      s_bi[wid * AROWS + row1] = bi1[j];
    }
  }
  __syncthreads();

  // Combine across the 16 waves; thread t<32 owns row t.
  if (tid < AROWS) {
    float v = s_bv[tid]; int idx = s_bi[tid];
    for (int w = 1; w < AWAVES; ++w) {
      const float ov = s_bv[w * AROWS + tid];
      const int   oi = s_bi[w * AROWS + tid];
      if (ov > v || (ov == v && oi < idx)) { v = ov; idx = oi; }
    }
    s_idx[tid] = idx;
    s_val[tid] = v;
    out_idx[r0 + tid] = (float)idx;
    atomicAdd(&counts[idx], 1u);
  }
  __syncthreads();

  if (tid == 0) {
    float s = 0.f;
    for (int r = 0; r < AROWS; ++r) s += s_val[r];
    atomicAdd(&scal[1], (double)s);
  }

  // Gather quantized rows + squared-error partial (16 threads per row).
  {
    const int row = tid >> 4;
    const int seg = (tid & 15) * 16;
    const int idx = s_idx[row];
    const float* q = codebook + (size_t)idx * DDIM + seg;
    const float* l = latent + (size_t)(r0 + row) * DDIM + seg;
    float* o = out_q + (size_t)(r0 + row) * DDIM + seg;
    float s = 0.f;
#pragma unroll
    for (int e = 0; e < 16; ++e) {
      const float qq = q[e];
      const float dd = l[e] - qq;
      o[e] = qq;
      s += dd * dd;
    }
    s_red[tid] = s;
  }
  __syncthreads();
  for (int st = ATHREADS / 2; st > 0; st >>= 1) {
    if (tid < st) s_red[tid] += s_red[tid + st];
    __syncthreads();
  }
  if (tid == 0) atomicAdd(&scal[0], (double)s_red[0]);
}

// ---------------------------------------------------------------------------
// Codebook pairwise distances: one 32x32 macro-tile of C*C^T per wave, four
// independent WMMA accumulators (4 loads : 4 WMMA per K-step).
// Symmetry on macro-tiles: skip JT<IT; JT>IT counts x2 (mirror skipped);
// JT==IT counts x1 (its off-diagonal sub-tiles mirror each other in-tile).
// d = sqrt(max(sq_i + sq_j - 2*dot, 0)); accumulate off-diagonal sum + min.
__global__ __launch_bounds__(256)
void vq_pairwise(const float* __restrict__ cb, const float* __restrict__ sq,
                 double* __restrict__ scal, unsigned* __restrict__ minbits) {
  const int tid  = threadIdx.x;
  const int wid  = tid >> 5;
  const int lane = tid & 31;
  const int half = lane >> 4;
  const int l16  = lane & 15;

  const int MT   = KCODES / 32;                // 128 macro-tiles per side
  const int tile = blockIdx.x * 8 + wid;       // 128*128 macro-tiles total
  const int IT = tile / MT;
  const int JT = tile % MT;

  float lsum = 0.f;
  float lmin = 3.4e38f;

  if (JT >= IT) {                              // wave-uniform branch
    const int i0 = IT * 32, j0 = JT * 32;
    const float* a0 = cb + (size_t)(i0 + l16) * DDIM + 2 * half;
    const float* a1 = a0 + 16 * DDIM;
    const float* b0 = cb + (size_t)(j0 + l16) * DDIM + 2 * half;
    const float* b1 = b0 + 16 * DDIM;

    v8f c00 = {}, c01 = {}, c10 = {}, c11 = {};
#pragma unroll 4
    for (int kk = 0; kk < 64; ++kk) {
      v2f xa0 = *(const v2f*)(a0 + 4 * kk);
      v2f xa1 = *(const v2f*)(a1 + 4 * kk);
      v2f xb0 = *(const v2f*)(b0 + 4 * kk);
      v2f xb1 = *(const v2f*)(b1 + 4 * kk);
      c00 = __builtin_amdgcn_wmma_f32_16x16x4_f32(false, xa0, false, xb0, (short)0,
                                                  c00, false, false);
      c01 = __builtin_amdgcn_wmma_f32_16x16x4_f32(false, xa0, false, xb1, (short)0,
                                                  c01, false, false);
      c10 = __builtin_amdgcn_wmma_f32_16x16x4_f32(false, xa1, false, xb0, (short)0,
                                                  c10, false, false);
      c11 = __builtin_amdgcn_wmma_f32_16x16x4_f32(false, xa1, false, xb1, (short)0,
                                                  c11, false, false);
    }

    const float sqc0 = sq[j0 + l16];
    const float sqc1 = sq[j0 + 16 + l16];
#pragma unroll
    for (int j = 0; j < 8; ++j) {
      const int gr0 = i0 + j + 8 * half;
      const int gr1 = gr0 + 16;
      const float sqr0 = sq[gr0];
      const float sqr1 = sq[gr1];
      const int gc0 = j0 + l16;
      const int gc1 = gc0 + 16;

      float d = sqrtf(fmaxf(sqr0 + sqc0 - 2.0f * c00[j], 0.0f));
      if (gr0 != gc0) { lsum += d; lmin = fminf(lmin, d); }
      d = sqrtf(fmaxf(sqr0 + sqc1 - 2.0f * c01[j], 0.0f));
      if (gr0 != gc1) { lsum += d; lmin = fminf(lmin, d); }
      d = sqrtf(fmaxf(sqr1 + sqc0 - 2.0f * c10[j], 0.0f));
      if (gr1 != gc0) { lsum += d; lmin = fminf(lmin, d); }
      d = sqrtf(fmaxf(sqr1 + sqc1 - 2.0f * c11[j], 0.0f));
      if (gr1 != gc1) { lsum += d; lmin = fminf(lmin, d); }
    }
#pragma unroll
    for (int off = 16; off >= 1; off >>= 1) {
      lsum += __shfl_xor(lsum, off, 32);
      lmin = fminf(lmin, __shfl_xor(lmin, off, 32));
    }
    if (JT > IT) lsum *= 2.0f;                 // mirror macro-tile skipped
  }

  __shared__ float s_sum[8];
  __shared__ float s_min[8];
  if (lane == 0) { s_sum[wid] = lsum; s_min[wid] = lmin; }
  __syncthreads();
  if (tid == 0) {
    float ts = 0.f, tm = 3.4e38f;
    for (int w = 0; w < 8; ++w) { ts += s_sum[w]; tm = fminf(tm, s_min[w]); }
    atomicAdd(&scal[2], (double)ts);
    atomicMin(minbits, __float_as_uint(tm));
  }
}

// ---------------------------------------------------------------------------
__global__ __launch_bounds__(256)
void vq_finalize(const unsigned* __restrict__ counts, const double* __restrict__ scal,
                 const unsigned* __restrict__ minbits, float* __restrict__ out_s) {
  __shared__ double sh[256];
  const int tid = threadIdx.x;
  double h = 0.0;
  for (int k = tid; k < KCODES; k += 256) {
    const double p = (double)counts[k] / (double)N_TOK;
    h -= p * log(p + 1e-10);
  }
  sh[tid] = h;
  __syncthreads();
  for (int st = 128; st > 0; st >>= 1) {
    if (tid < st) sh[tid] += sh[tid + st];
    __syncthreads();
  }
  if (tid == 0) {
    const double mse = scal[0] / ((double)N_TOK * (double)DDIM);
    out_s[0] = (float)(0.25 * mse);                                   // commitment
    out_s[1] = (float)mse;                                            // codebook loss
    out_s[2] = (float)exp(sh[0]);                                     // perplexity
    out_s[3] = (float)(scal[1] / (double)N_TOK);                      // selected cos
    out_s[4] = (float)(scal[2] / ((double)KCODES * (KCODES - 1)));    // avg euclid
    out_s[5] = __uint_as_float(minbits[0]);                           // min euclid
  }
}

// ---------------------------------------------------------------------------
extern "C" void kernel_launch(void* const* d_in, const int* in_sizes, int n_in,
                              void* d_out, int out_size, void* d_ws, size_t ws_size,
                              hipStream_t stream) {
  (void)in_sizes; (void)n_in; (void)out_size; (void)ws_size;
  const float* latent   = (const float*)d_in[0];
  const float* codebook = (const float*)d_in[1];

  float* out       = (float*)d_out;
  float* out_q     = out;                                  // [N_TOK*DDIM]
  float* out_idx   = out + (size_t)N_TOK * DDIM;           // [N_TOK]
  float* out_scal  = out_idx + N_TOK;                      // [6]

  // Workspace layout (~12.6 MB): ln | cn | sq | counts | 3 doubles | minbits.
  float*    ln      = (float*)d_ws;
  float*    cn      = ln + (size_t)N_TOK * DDIM;
  float*    sq      = cn + (size_t)KCODES * DDIM;
  unsigned* counts  = (unsigned*)(sq + KCODES);
  double*   scal    = (double*)(counts + KCODES);          // offset is 8B aligned
  unsigned* minbits = (unsigned*)(scal + 3);

  vq_init<<<(KCODES + 255) / 256, 256, 0, stream>>>(counts, scal, minbits);
  vq_rownorm<<<N_TOK / 8, 256, 0, stream>>>(latent, ln, nullptr, N_TOK);
  vq_rownorm<<<KCODES / 8, 256, 0, stream>>>(codebook, cn, sq, KCODES);
  vq_assign<<<N_TOK / AROWS, ATHREADS, 0, stream>>>(latent, codebook, ln, cn, counts,
                                                    out_q, out_idx, scal);
  {
    const int MT = KCODES / 32;
    vq_pairwise<<<(MT * MT) / 8, 256, 0, stream>>>(codebook, sq, scal, minbits);
  }
  vq_finalize<<<1, 256, 0, stream>>>(counts, scal, minbits, out_scal);
}